// TorchPipeline_66400194396291
// MI455X (gfx1250) — compile-verified
//
#include <hip/hip_runtime.h>

// ---------------------------------------------------------------------------
// Problem constants (reference: B=2048, C=8, T=1024)
// ---------------------------------------------------------------------------
#define BB 2048
#define TT 1024
#define NELEM (BB * 8 * TT)            // 16,777,216 elements of x
#define INV_N (1.0f / 2097152.0f)      // 1/(B*T)
#define EPS 1e-5f
#define SLOPE (1.0f / 128.0f)

typedef __attribute__((ext_vector_type(2))) float v2f;
typedef __attribute__((ext_vector_type(8))) float v8f;

__device__ __forceinline__ v8f wmma4(v2f a, v2f b, v8f c) {
    // D = A(16x4 f32) x B(4x16 f32) + C(16x16 f32)
    return __builtin_amdgcn_wmma_f32_16x16x4_f32(
        /*neg_a=*/false, a, /*neg_b=*/false, b,
        /*c_mod=*/(short)0, c, /*reuse_a=*/false, /*reuse_b=*/false);
}

__device__ __forceinline__ float leaky(float x) {
    return x >= 0.f ? x : x * SLOPE;
}

// ---------------------------------------------------------------------------
// K1: per-channel sum / sumsq of x, deterministic per-block tree reduction.
// Each thread handles 16 consecutive elements (one channel). Each 64-thread
// group of a block shares one channel: c = (blockIdx*4 + tid/64) & 7.
// ---------------------------------------------------------------------------
__global__ __launch_bounds__(256) void stats1_partial(const float* __restrict__ X,
                                                      float* __restrict__ part1) {
    __shared__ float sS[256], sQ[256];
    int tid = threadIdx.x;
    size_t g = (size_t)blockIdx.x * 256 + tid;
    size_t base = g * 16;
    float s = 0.f, q = 0.f;
#pragma unroll
    for (int k = 0; k < 4; ++k) {
        float4 v = *(const float4*)(X + base + k * 4);
        s += v.x + v.y + v.z + v.w;
        q += v.x * v.x + v.y * v.y + v.z * v.z + v.w * v.w;
    }
    sS[tid] = s;
    sQ[tid] = q;
    __syncthreads();
#pragma unroll
    for (int off = 32; off >= 1; off >>= 1) {
        if ((tid & 63) < off) {
            sS[tid] += sS[tid + off];
            sQ[tid] += sQ[tid + off];
        }
        __syncthreads();
    }
    if ((tid & 63) == 0) {
        int grp = tid >> 6;
        part1[(size_t)blockIdx.x * 8 + grp * 2 + 0] = sS[tid];
        part1[(size_t)blockIdx.x * 8 + grp * 2 + 1] = sQ[tid];
    }
}

__global__ void stats1_reduce(const float* __restrict__ part1, float* __restrict__ w) {
    int tid = threadIdx.x;
    if (tid < 16) {
        int c = tid >> 1, st = tid & 1;
        float acc = 0.f;
        for (int k = 0; k < 2048; ++k) {
            int blk = (c < 4) ? 2 * k : 2 * k + 1;  // even blocks: ch0-3, odd: ch4-7
            acc += part1[(size_t)blk * 8 + (c & 3) * 2 + st];
        }
        w[2 * c + st] = acc;
    }
}

// ---------------------------------------------------------------------------
// K2: reservoir recurrence. One lane per batch element. A is uniform ->
// scalar loads; f state + accumulators in VGPRs; stores f[b][c][t] (fp32).
// ---------------------------------------------------------------------------
__global__ __launch_bounds__(256) void reservoir_kernel(
    const float* __restrict__ X, const float* __restrict__ Ap,
    const float* __restrict__ g1, const float* __restrict__ bb1,
    const float* __restrict__ ws, float* __restrict__ fbuf) {
    int b = blockIdx.x * 256 + threadIdx.x;
    float Am[8][8];
#pragma unroll
    for (int i = 0; i < 8; ++i)
#pragma unroll
        for (int j = 0; j < 8; ++j) Am[i][j] = Ap[i * 8 + j];

    float sc[8], sh[8];
#pragma unroll
    for (int c = 0; c < 8; ++c) {
        float m = ws[2 * c] * INV_N;
        float v = ws[2 * c + 1] * INV_N - m * m;
        float s = g1[c] * rsqrtf(v + EPS);
        sc[c] = s;
        sh[c] = bb1[c] - m * s;
    }
    float fst[8];
#pragma unroll
    for (int i = 0; i < 8; ++i) fst[i] = 0.f;

    const size_t rowB = (size_t)b * 8 * TT;
    for (int t = 0; t < TT; t += 4) {
        float4 xv[8], fo[8];
#pragma unroll
        for (int c = 0; c < 8; ++c) xv[c] = *(const float4*)(X + rowB + c * TT + t);
#pragma unroll
        for (int tt = 0; tt < 4; ++tt) {
            float fnew[8];
#pragma unroll
            for (int i = 0; i < 8; ++i) {
                float comp = (tt == 0) ? xv[i].x : (tt == 1) ? xv[i].y : (tt == 2) ? xv[i].z : xv[i].w;
                float a = comp * sc[i] + sh[i];
#pragma unroll
                for (int j = 0; j < 8; ++j) a = fmaf(Am[i][j], fst[j], a);
                a *= 4.f;
                a = fminf(1.f, fmaxf(-1.f, a));
                fnew[i] = a;
            }
#pragma unroll
            for (int i = 0; i < 8; ++i) {
                fst[i] = fnew[i];
                if (tt == 0) fo[i].x = fnew[i];
                else if (tt == 1) fo[i].y = fnew[i];
                else if (tt == 2) fo[i].z = fnew[i];
                else fo[i].w = fnew[i];
            }
        }
#pragma unroll
        for (int c = 0; c < 8; ++c) *(float4*)(fbuf + rowB + c * TT + t) = fo[c];
    }
}

// ---------------------------------------------------------------------------
// K3/K5: fused feed-forward + stacked GEMM via V_WMMA_F32_16X16X4_F32.
// One wave per b; 64 tiles of 16 columns (t). A = activations (M=16 cols,
// K=4 in-channels), B = weights (K=4 in-ch, N=16 out-ch), D = [cols, outch].
// FINAL=0: accumulate bn2 sum/sumsq partials per b.
// FINAL=1: apply bn2 + leaky, mean over T, write d_out.
// ---------------------------------------------------------------------------
template <int FINAL>
__global__ __launch_bounds__(256) void fused_gemm(
    const float* __restrict__ X,
    const float* __restrict__ w1, const float* __restrict__ b1,
    const float* __restrict__ w2, const float* __restrict__ b2,
    const float* __restrict__ w3, const float* __restrict__ b3,
    const float* __restrict__ g1, const float* __restrict__ bb1,
    const float* __restrict__ g2, const float* __restrict__ bb2,
    const float* __restrict__ ws, const float* __restrict__ fbuf,
    float* __restrict__ part2, float* __restrict__ outp) {
    __shared__ float w1B[8 * 33];       // [k][n], n<24 valid (padded, stride 33)
    __shared__ float w2B[24 * 16];      // [k][n]
    __shared__ float w3B[48 * 16];      // [k][n], n<11 valid
    __shared__ float bias1[32], bias2s[16], bias3s[16];
    __shared__ float stk[8][48 * 17];   // per-wave stacked [ch][col], stride 17
    __shared__ float hS[8][32 * 17];    // per-wave h tile   [ch][col]

    int tid = threadIdx.x;
    for (int i = tid; i < 1472; i += 256) {
        if (i < 256) {
            int k = i >> 5, n = i & 31;
            w1B[k * 33 + n] = (n < 24) ? w1[n * 8 + k] : 0.f;
        } else if (i < 640) {
            int j = i - 256;
            int k = j >> 4, n = j & 15;
            w2B[k * 16 + n] = w2[n * 24 + k];
        } else if (i < 1408) {
            int j = i - 640;
            int k = j >> 4, n = j & 15;
            w3B[k * 16 + n] = (n < 11) ? w3[n * 48 + k] : 0.f;
        } else {
            int j = i - 1408;
            if (j < 32) bias1[j] = (j < 24) ? b1[j] : 0.f;
            else if (j < 48) bias2s[j - 32] = b2[j - 32];
            else bias3s[j - 48] = (j < 59) ? b3[j - 48] : 0.f;
        }
    }
    __syncthreads();

    const int lane = tid & 31;
    const int wv = tid >> 5;
    const int colN = lane & 15;    // N index (out-channel) / column for loads
    const int cHalf = lane >> 4;
    const int klo = cHalf * 2;     // A/B operand K interleave: V0 = K0|K2, V1 = K1|K3
    const int b = blockIdx.x * 8 + wv;

    // bn1 per-lane scale/shift for channels c = 2*cc + cHalf
    float scL[4], shL[4];
#pragma unroll
    for (int cc = 0; cc < 4; ++cc) {
        int c = cc * 2 + cHalf;
        float m = ws[2 * c] * INV_N;
        float v = ws[2 * c + 1] * INV_N - m * m;
        float s = g1[c] * rsqrtf(v + EPS);
        scL[cc] = s;
        shL[cc] = bb1[c] - m * s;
    }
    float b1Lo = bias1[colN], b1Hi = bias1[16 + colN];
    float b2L = bias2s[colN], b3L = bias3s[colN];

    float a2 = 0.f, c2c = 0.f;
    if (FINAL) {
        int chc = colN < 11 ? colN : 10;
        float m2 = ws[16 + 2 * chc] * INV_N;
        float v2 = ws[16 + 2 * chc + 1] * INV_N - m2 * m2;
        float rs = rsqrtf(v2 + EPS);
        float gg = colN < 11 ? g2[chc] : 0.f;
        float bv = colN < 11 ? bb2[chc] : 0.f;
        a2 = rs * gg;
        c2c = bv - m2 * a2;
    }

    float* st = stk[wv];
    float* hh = hS[wv];
    const size_t rowB = (size_t)b * 8192;
    const v8f zero8 = {0.f, 0.f, 0.f, 0.f, 0.f, 0.f, 0.f, 0.f};

    float s1 = 0.f, s2 = 0.f, accY = 0.f;

    for (int t0 = 0; t0 < TT; t0 += 16) {
        // ---- stage x, bw, f, leaky(f) into LDS (wave-local region) ----
#pragma unroll
        for (int cc = 0; cc < 4; ++cc) {
            int c = cc * 2 + cHalf;
            float xv = X[rowB + c * TT + t0 + colN];
            st[c * 17 + colN] = xv;                         // stacked rows 0-7 : x
            st[(8 + c) * 17 + colN] = xv * scL[cc] + shL[cc];  // rows 8-15 : bw
            float fv = fbuf[rowB + c * TT + t0 + colN];
            st[(32 + c) * 17 + colN] = fv;                  // rows 32-39: f
            st[(40 + c) * 17 + colN] = leaky(fv);           // rows 40-47: leaky(f)
        }
        // ---- layer 1: h = leaky(W1 * bw + b1), 24 out-ch in 2 N-tiles ----
#pragma unroll
        for (int ht = 0; ht < 2; ++ht) {
            v8f acc = zero8;
#pragma unroll
            for (int kc = 0; kc < 8; kc += 4) {
                v2f av, bv;
                av.x = st[(8 + kc + klo) * 17 + colN];
                av.y = st[(8 + kc + klo + 1) * 17 + colN];
                bv.x = w1B[(kc + klo) * 33 + ht * 16 + colN];
                bv.y = w1B[(kc + klo + 1) * 33 + ht * 16 + colN];
                acc = wmma4(av, bv, acc);
            }
            float bs = ht ? b1Hi : b1Lo;
#pragma unroll
            for (int r = 0; r < 8; ++r) {
                float hv = leaky(acc[r] + bs);
                hh[(ht * 16 + colN) * 17 + r + 8 * cHalf] = hv;
            }
        }
        // ---- layer 2: ff = leaky(W2 * h + b2) -> stacked rows 16-31 ----
        {
            v8f acc = zero8;
#pragma unroll
            for (int kc = 0; kc < 24; kc += 4) {
                v2f av, bv;
                av.x = hh[(kc + klo) * 17 + colN];
                av.y = hh[(kc + klo + 1) * 17 + colN];
                bv.x = w2B[(kc + klo) * 16 + colN];
                bv.y = w2B[(kc + klo + 1) * 16 + colN];
                acc = wmma4(av, bv, acc);
            }
#pragma unroll
            for (int r = 0; r < 8; ++r) {
                float fv = leaky(acc[r] + b2L);
                st[(16 + colN) * 17 + r + 8 * cHalf] = fv;
            }
        }
        // ---- layer 3: pre = W3 * stacked + b3 (K=48) ----
        {
            v8f acc = zero8;
#pragma unroll
            for (int kc = 0; kc < 48; kc += 4) {
                v2f av, bv;
                av.x = st[(kc + klo) * 17 + colN];
                av.y = st[(kc + klo + 1) * 17 + colN];
                bv.x = w3B[(kc + klo) * 16 + colN];
                bv.y = w3B[(kc + klo + 1) * 16 + colN];
                acc = wmma4(av, bv, acc);
            }
#pragma unroll
            for (int r = 0; r < 8; ++r) {
                float p = acc[r] + b3L;
                if (FINAL) {
                    accY += leaky(p * a2 + c2c);
                } else {
                    s1 += p;
                    s2 += p * p;
                }
            }
        }
    }

    if (FINAL) {
        accY += __shfl_xor(accY, 16, 32);
        if (lane < 11) outp[b * 11 + lane] = accY * (1.f / 1024.f);
    } else {
        s1 += __shfl_xor(s1, 16, 32);
        s2 += __shfl_xor(s2, 16, 32);
        if (lane < 11) {
            part2[(size_t)b * 32 + 2 * lane + 0] = s1;
            part2[(size_t)b * 32 + 2 * lane + 1] = s2;
        }
    }
}

// Deterministic serial reduce of bn2 partials: 22 threads, one (ch,stat) each.
__global__ void stats2_reduce(const float* __restrict__ part2, float* __restrict__ w) {
    int tid = threadIdx.x;
    if (tid < 22) {
        int ch = tid >> 1, st = tid & 1;
        float acc = 0.f;
        for (int b = 0; b < BB; ++b) acc += part2[(size_t)b * 32 + 2 * ch + st];
        w[16 + 2 * ch + st] = acc;
    }
}

// ---------------------------------------------------------------------------
// Launch. Workspace layout (floats):
//  [0..15]   bn1 sum/sumsq  [16..47] bn2 sum/sumsq  [48..63] spare
//  [64 .. 64+16M)            f state [B][8][T]        (64 MiB)
//  then 4096*8 part1, then 2048*32 part2              (~0.4 MiB)
// ---------------------------------------------------------------------------
extern "C" void kernel_launch(void* const* d_in, const int* in_sizes, int n_in,
                              void* d_out, int out_size, void* d_ws, size_t ws_size,
                              hipStream_t stream) {
    const float* X = (const float*)d_in[0];
    const float* Ap = (const float*)d_in[1];
    const float* g1 = (const float*)d_in[2];
    const float* bb1 = (const float*)d_in[3];
    const float* w1 = (const float*)d_in[4];
    const float* b1 = (const float*)d_in[5];
    const float* w2 = (const float*)d_in[6];
    const float* b2 = (const float*)d_in[7];
    const float* w3 = (const float*)d_in[8];
    const float* b3 = (const float*)d_in[9];
    const float* g2 = (const float*)d_in[10];
    const float* bb2 = (const float*)d_in[11];

    float* w = (float*)d_ws;
    float* fbuf = w + 64;
    float* part1 = fbuf + (size_t)NELEM;
    float* part2 = part1 + 4096 * 8;
    float* outp = (float*)d_out;

    stats1_partial<<<4096, 256, 0, stream>>>(X, part1);
    stats1_reduce<<<1, 32, 0, stream>>>(part1, w);
    reservoir_kernel<<<8, 256, 0, stream>>>(X, Ap, g1, bb1, w, fbuf);
    fused_gemm<0><<<256, 256, 0, stream>>>(X, w1, b1, w2, b2, w3, b3, g1, bb1,
                                           g2, bb2, w, fbuf, part2, outp);
    stats2_reduce<<<1, 64, 0, stream>>>(part2, w);
    fused_gemm<1><<<256, 256, 0, stream>>>(X, w1, b1, w2, b2, w3, b3, g1, bb1,
                                           g2, bb2, w, fbuf, part2, outp);
}